// Module1_11879879541811
// MI455X (gfx1250) — compile-verified
//
#include <hip/hip_runtime.h>
#include <stdint.h>

// Elementwise: out[i] = float( x*2 if x in NUMS else x ), x in [0,64).
// NUMS = {3..9, 14..18, 22..31, 37..39, 46..57} -> 64-bit membership bitmask
// (37 bits set == |NUMS|).
static constexpr uint64_t kMask = 0x03FFC0E0FFC7C3F8ull;

typedef int   __attribute__((ext_vector_type(4))) vint4;
typedef float __attribute__((ext_vector_type(4))) vfloat4;

#define TPB      256
#define NBLOCKS  2048
#define DEPTH    4   // LDS ring slots; pipeline keeps 3 async copies in flight

__global__ __launch_bounds__(TPB) void member_double_kernel(const int* __restrict__ in,
                                                            float* __restrict__ out,
                                                            long long n) {
  // 4-slot LDS staging ring: 4 x 256 x 16B = 16 KB (of 320 KB/WGP).
  __shared__ vint4 tile[DEPTH][TPB];

  const int      tid     = threadIdx.x;
  const uint64_t n4      = (uint64_t)n >> 2;                 // # of int4 vectors
  const uint64_t stride4 = (uint64_t)gridDim.x * TPB;
  uint64_t       idx4    = (uint64_t)blockIdx.x * TPB + tid;

  if (n4 != 0) {
    const uint64_t last4 = n4 - 1;

    // LDS byte offsets of this lane's ring slots: low 32 bits of the generic
    // address of a __shared__ object are the workgroup-relative LDS offset,
    // which is what GLOBAL_LOAD_ASYNC_TO_LDS's VDST operand expects.
    uint32_t lds[DEPTH];
#pragma unroll
    for (int j = 0; j < DEPTH; ++j)
      lds[j] = (uint32_t)(uintptr_t)&tile[j][tid];

    // Prologue: fill the pipeline with 3 async copies. Addresses are clamped
    // to the last valid vector so every lane/wave issues the same count
    // (redundant 16B loads are harmless; stores stay predicated).
#pragma unroll
    for (int j = 0; j < DEPTH - 1; ++j) {
      uint64_t a = idx4 + (uint64_t)j * stride4;
      if (a > last4) a = last4;
      uint32_t goff = (uint32_t)(a << 4);   // byte offset, < 2^31 here
      asm volatile("global_load_async_to_lds_b128 %0, %1, %2"
                   :: "v"(lds[j]), "v"(goff), "s"(in) : "memory");
    }

    int buf = 0;
    while (idx4 < n4) {
      // Issue the copy 3 iterations ahead into the slot we finished last
      // iteration (its ds_load result was already consumed -> no WAR hazard).
      uint64_t pre = idx4 + (uint64_t)(DEPTH - 1) * stride4;
      if (pre > last4) pre = last4;
      uint32_t goff = (uint32_t)(pre << 4);
      asm volatile("global_load_async_to_lds_b128 %0, %1, %2"
                   :: "v"(lds[(buf + DEPTH - 1) & (DEPTH - 1)]), "v"(goff), "s"(in)
                   : "memory");

      // Retire the oldest copy (async loads complete in order): allow 3
      // outstanding -> the current slot's data is now resident in LDS.
      asm volatile("s_wait_asynccnt 0x3" ::: "memory");

      // Each lane reads back only its own slot -> per-wave ASYNCcnt suffices,
      // no workgroup barrier needed. Lowers to ds_load_b128.
      vint4 v = tile[buf][tid];

      vfloat4 f;
#pragma unroll
      for (int k = 0; k < 4; ++k) {
        int x  = v[k];
        int sh = (int)((kMask >> (x & 63)) & 1);  // 1 if member, else 0
        f[k]   = (float)(x << sh);                // x*2 or x
      }

      // Streaming B128 store with non-temporal hint
      // (268 MB working set > 192 MB L2 -> don't thrash the cache).
      __builtin_nontemporal_store(f, (vfloat4*)out + idx4);

      idx4 += stride4;
      buf = (buf + 1) & (DEPTH - 1);
    }
  }

  // Scalar tail (n is divisible by 4 for this problem, but stay correct).
  const uint64_t rem = (uint64_t)n & 3ull;
  if (blockIdx.x == 0 && (uint64_t)tid < rem) {
    uint64_t i = (uint64_t)n - rem + tid;
    int x  = in[i];
    int sh = (int)((kMask >> (x & 63)) & 1);
    out[i] = (float)(x << sh);
  }
}

extern "C" void kernel_launch(void* const* d_in, const int* in_sizes, int n_in,
                              void* d_out, int out_size, void* d_ws, size_t ws_size,
                              hipStream_t stream) {
  (void)n_in; (void)d_ws; (void)ws_size; (void)out_size;
  const int* in  = (const int*)d_in[0];
  float*     out = (float*)d_out;
  long long  n   = (long long)in_sizes[0];  // 4096*8192 = 33,554,432

  long long n4     = (n + 3) >> 2;
  long long blocks = (n4 + TPB - 1) / TPB;
  if (blocks > NBLOCKS) blocks = NBLOCKS;
  if (blocks < 1) blocks = 1;

  member_double_kernel<<<dim3((unsigned)blocks), dim3(TPB), 0, stream>>>(in, out, n);
}